// BSpline1D_7395933684405
// MI455X (gfx1250) — compile-verified
//
#include <hip/hip_runtime.h>

// ---------------------------------------------------------------------------
// Cubic B-spline (uniform knots, h = 0.25 exact) on MI455X / gfx1250.
//
//  * Memory-bound: 32 MB traffic -> ~1.4 us floor at 23.3 TB/s.
//  * Closed-form uniform-cubic weights (de Boor collapses for uniform knots).
//  * Async global->LDS b128 staging (ASYNCcnt), double-buffered per wave.
//  * V_WMMA_F32_16X16X4_F32 contracts weights(16x4) x coeffs-gather(4x16);
//    per-point spline = diagonal D[p,p]. WMMA co-executes with VALU (TRANS).
// ---------------------------------------------------------------------------

typedef float v2f __attribute__((ext_vector_type(2)));
typedef float v8f __attribute__((ext_vector_type(8)));

#define WAVES_PER_BLOCK 8
#define PTS_PER_TILE    128                      // one async b128 per wave fills a tile
#define PTS_PER_WAVE    256                      // 2 tiles, double-buffered
#define PTS_PER_BLOCK   (WAVES_PER_BLOCK * PTS_PER_WAVE)   // 2048
#define COEFF_SLOTS     16                       // 11 coeffs, padded to 64 B

// Select d[k] (k in 0..7, per-lane) via 7 v_cndmask ops.
__device__ __forceinline__ float pick8(v8f d, int k) {
  float t0 = (k & 1) ? d[1] : d[0];
  float t1 = (k & 1) ? d[3] : d[2];
  float t2 = (k & 1) ? d[5] : d[4];
  float t3 = (k & 1) ? d[7] : d[6];
  float u0 = (k & 2) ? t1 : t0;
  float u1 = (k & 2) ? t3 : t2;
  return (k & 4) ? u1 : u0;
}

__global__ __launch_bounds__(256) void bspline1d_kernel(
    const float* __restrict__ x, const float* __restrict__ coeffs,
    const float* __restrict__ lwp, const float* __restrict__ biasp,
    float* __restrict__ out, int n) {
  __shared__ __align__(16) float smem[COEFF_SLOTS + WAVES_PER_BLOCK * PTS_PER_WAVE];

  const int tid = threadIdx.x;
  if (tid < 11) smem[tid] = coeffs[tid];
  __syncthreads();

  const float lw   = lwp[0];
  const float bias = biasp[0];

  const int  wave = tid >> 5;
  const int  lane = tid & 31;
  const bool lo   = lane < 16;

  const long long waveBase =
      ((long long)blockIdx.x * WAVES_PER_BLOCK + wave) * PTS_PER_WAVE;
  if (waveBase >= n) return;

  float* buf = &smem[COEFF_SLOTS + wave * PTS_PER_WAVE];

  // Issue both tiles' async global->LDS copies up front (ASYNCcnt = 2).
  // Per lane: 16 B, tile = 32 lanes * 16 B = 512 B contiguous.
#pragma unroll
  for (int t = 0; t < 2; ++t) {
    unsigned ldsAddr =
        (unsigned)(size_t)(buf + t * PTS_PER_TILE) + (unsigned)lane * 16u;
    const float* g = x + waveBase + t * PTS_PER_TILE + lane * 4;
    asm volatile("global_load_async_to_lds_b128 %0, %1, off"
                 :: "v"(ldsAddr), "v"(g) : "memory");
  }

#pragma unroll
  for (int t = 0; t < 2; ++t) {
    if (t == 0) asm volatile("s_wait_asynccnt 0x1" ::: "memory");
    else        asm volatile("s_wait_asynccnt 0x0" ::: "memory");

    const float*    tb    = buf + t * PTS_PER_TILE;
    const long long tbase = waveBase + t * PTS_PER_TILE;

#pragma unroll
    for (int g = 0; g < 4; ++g) {          // 32 points per group, 1 per lane
      float xv = tb[g * 32 + lane];        // ds_load from async-staged tile

      // interval + local parameter: t = (x + 1.75) * 4 = 4x + 7 (exact knots)
      float tt = fmaf(xv, 4.0f, 7.0f);
      float sf = floorf(tt);
      float u  = tt - sf;
      int   j0 = (int)sf - 3;              // first nonzero basis index
      j0 = min(max(j0, 0), 7);             // keep gathers in [0,10]

      // uniform cubic B-spline weights (partition of unity)
      float um  = 1.0f - u;
      float u2  = u * u;
      float um2 = um * um;
      float w0  = (1.0f / 6.0f) * um2 * um;
      float w3  = (1.0f / 6.0f) * u2 * u;
      float w1  = fmaf(u2,  fmaf(0.5f, u,  -1.0f), 2.0f / 3.0f);
      float w2  = fmaf(um2, fmaf(0.5f, um, -1.0f), 2.0f / 3.0f);

      // lane-half swap (ds_swizzle SWAPX16) for WMMA A/B packing
      float xs  = __shfl_xor(xv, 16);
      float w0s = __shfl_xor(w0, 16);
      float w1s = __shfl_xor(w1, 16);
      float w2s = __shfl_xor(w2, 16);
      float w3s = __shfl_xor(w3, 16);
      int   j0s = __shfl_xor(j0, 16);

      // A (16x4 f32): VGPR0 = K0 (lanes 0-15) | K2 (lanes 16-31); VGPR1 = K1 | K3
      v2f A1 = { lo ? w0  : w2s, lo ? w1  : w3s };   // points g*32 + 0..15
      v2f A2 = { lo ? w0s : w2,  lo ? w1s : w3  };   // points g*32 + 16..31

      // B (4x16 f32): column n = gathered coeffs of point n
      int bi1 = (lo ? j0  : j0s) + (lo ? 0 : 2);
      int bi2 = (lo ? j0s : j0 ) + (lo ? 0 : 2);
      v2f B1 = { smem[bi1], smem[bi1 + 1] };
      v2f B2 = { smem[bi2], smem[bi2 + 1] };

      v8f Z = {};
      v8f D1 = __builtin_amdgcn_wmma_f32_16x16x4_f32(
          false, A1, false, B1, (short)0, Z, false, false);
      v8f D2 = __builtin_amdgcn_wmma_f32_16x16x4_f32(
          false, A2, false, B2, (short)0, Z, false, false);

      // Diagonal lives on lanes 0-7 (vgpr = lane) and 24-31 (vgpr = lane-24).
      int   k  = lane & 7;
      float s1 = pick8(D1, k);
      float s2 = pick8(D2, k);

      if (lane < 8 || lane >= 24) {
        float xl1 = lo ? xv : xs;   // x of point (lane & 15)
        float xl2 = lo ? xs : xv;   // x of point 16 + (lane & 15)
        long long p = tbase + g * 32 + (lane & 15);
        out[p]      = fmaf(lw, xl1, bias) + s1;
        out[p + 16] = fmaf(lw, xl2, bias) + s2;
      }
    }
  }
}

extern "C" void kernel_launch(void* const* d_in, const int* in_sizes, int n_in,
                              void* d_out, int out_size, void* d_ws, size_t ws_size,
                              hipStream_t stream) {
  (void)n_in; (void)out_size; (void)d_ws; (void)ws_size;
  const float* x      = (const float*)d_in[0];
  const float* coeffs = (const float*)d_in[1];
  const float* lw     = (const float*)d_in[2];
  const float* bias   = (const float*)d_in[3];
  float*       out    = (float*)d_out;
  int n = in_sizes[0];                       // 4194304, multiple of 2048
  int blocks = (n + PTS_PER_BLOCK - 1) / PTS_PER_BLOCK;
  hipLaunchKernelGGL(bspline1d_kernel, dim3(blocks), dim3(256), 0, stream,
                     x, coeffs, lw, bias, out, n);
}